// FusiformerLayer_24979529793595
// MI455X (gfx1250) — compile-verified
//
#include <hip/hip_runtime.h>
#include <math.h>

#define DIM_  256
#define H_    8
#define HD_   32
#define HID_  512
#define SCALE_ 0.17677669529663687f   // 32^-0.5
#define APITCH 260                    // 16x256 A tile pitch: lm*260%64 = lm*4 -> conflict-free, 8B aligned
#define HPITCH (HID_ + 4)             // hidden tile pitch (phase-2 A reads conflict-free)

typedef float v2f __attribute__((ext_vector_type(2)));
typedef float v8f __attribute__((ext_vector_type(8)));
typedef int   v2i __attribute__((ext_vector_type(2)));

static __device__ __forceinline__ v8f wmma4(v2f a, v2f b, v8f c) {
  // V_WMMA_F32_16X16X4_F32 : exact fp32 16x16 tile MAC, K=4 per issue
  return __builtin_amdgcn_wmma_f32_16x16x4_f32(false, a, false, b, (short)0, c, false, false);
}

static __device__ __forceinline__ float wave_sum(float v) {
#pragma unroll
  for (int m = 16; m >= 1; m >>= 1) v += __shfl_xor(v, m, 32);
  return v;
}

static __device__ __forceinline__ float gelu_f(float x) {
  return 0.5f * x * (1.0f + erff(x * 0.70710678118654752440f));
}

// order-preserving float<->uint mapping for atomicMax on floats (0u == minimum)
static __device__ __forceinline__ unsigned f2ord(float f) {
  unsigned u = __float_as_uint(f);
  return (u & 0x80000000u) ? ~u : (u | 0x80000000u);
}
static __device__ __forceinline__ float ord2f(unsigned u) {
  return (u & 0x80000000u) ? __uint_as_float(u & 0x7FFFFFFFu) : __uint_as_float(~u);
}

// ---- CDNA5 async global->LDS copy (ASYNCcnt path), with safe fallback ------
static __device__ __forceinline__ void async_ld_b64(float* lds_dst, const float* gsrc) {
#if __has_builtin(__builtin_amdgcn_global_load_async_to_lds_b64)
  __builtin_amdgcn_global_load_async_to_lds_b64(
      (__attribute__((address_space(1))) v2i*)(void*)gsrc,
      (__attribute__((address_space(3))) v2i*)(void*)lds_dst, 0, 0);
#else
  lds_dst[0] = gsrc[0];
  lds_dst[1] = gsrc[1];
#endif
}
static __device__ __forceinline__ void wait_async() {
#if __has_builtin(__builtin_amdgcn_s_wait_asynccnt)
  __builtin_amdgcn_s_wait_asynccnt(0);
#else
  asm volatile("s_wait_asynccnt 0" ::: "memory");
#endif
}

// ---------------------------------------------------------------------------
// fp32 WMMA GEMM, K fixed at 256: C[M x Nc] = act(A(+Aadd) @ B + bias) + resid
// One block covers all Nc columns of a 16-row strip: blockDim.x = (Nc/64)*32.
// The shared 16x256 A tile is staged into LDS once (async on CDNA5).
// ---------------------------------------------------------------------------
template<bool HASADD>
__global__ __launch_bounds__(384)
void gemm_wmma_f32(const float* __restrict__ A, const float* __restrict__ Aadd,
                   const float* __restrict__ B, const float* __restrict__ bias,
                   const float* __restrict__ resid, float* __restrict__ C,
                   int Nc, int act)
{
  __shared__ float As[16 * APITCH];
  const int tid   = threadIdx.x;
  const int nthr  = blockDim.x;
  const int tileM = blockIdx.x * 16;

  // stage A tile (16 rows x 256 cols) into LDS; 2048 8-byte chunks
  for (int c = tid; c < 2048; c += nthr) {
    const int f = c * 2;
    const int row = f >> 8;          // /256
    const int col = f & 255;
    const float* g = A + (size_t)(tileM + row) * DIM_ + col;
    if (HASADD) {
      const float* g2 = Aadd + (size_t)(tileM + row) * DIM_ + col;
      As[row * APITCH + col]     = g[0] + g2[0];
      As[row * APITCH + col + 1] = g[1] + g2[1];
    } else {
      async_ld_b64(&As[row * APITCH + col], g);
    }
  }
  if (!HASADD) wait_async();
  __syncthreads();

  const int lane  = tid & 31;
  const int wv    = tid >> 5;
  const int lm    = lane & 15;
  const int kh    = lane >> 4;          // lane-half selects K pair
  const int tileN = wv * 64;

  v8f acc[4];
#pragma unroll
  for (int t = 0; t < 4; ++t)
#pragma unroll
    for (int r = 0; r < 8; ++r) acc[t][r] = 0.0f;

  const float* Ap = &As[lm * APITCH + kh * 2];
  const float* Bp = B + (size_t)(kh * 2) * Nc + tileN + lm;
  const size_t bstep = (size_t)4 * Nc;

#pragma unroll 2
  for (int k = 0; k < DIM_; k += 4) {
    v2f a; a.x = Ap[0]; a.y = Ap[1];
#pragma unroll
    for (int t = 0; t < 4; ++t) {
      v2f b;
      b.x = Bp[t * 16];        // K = k0 row
      b.y = Bp[Nc + t * 16];   // K = k0+1 row
      acc[t] = wmma4(a, b, acc[t]);
    }
    Ap += 4;
    Bp += bstep;
  }

#pragma unroll
  for (int t = 0; t < 4; ++t) {
    const int col = tileN + t * 16 + lm;
    const float bc = bias ? bias[col] : 0.0f;
#pragma unroll
    for (int r = 0; r < 8; ++r) {
      const int row = tileM + r + kh * 8;
      float v = acc[t][r] + bc;
      if (act == 1)      v = gelu_f(v);
      else if (act == 2) v = tanhf(v);
      if (resid) v += resid[(size_t)row * Nc + col];
      C[(size_t)row * Nc + col] = v;
    }
  }
}

// ---------------------------------------------------------------------------
// Fused MLP: Y = X + gelu(X@w1 + b1) @ w2 + b2. X tile and the 16x512 hidden
// tile both live in LDS; hidden activations never touch HBM.
// ---------------------------------------------------------------------------
__global__ __launch_bounds__(256)
void mlp_fused(const float* __restrict__ X, const float* __restrict__ w1,
               const float* __restrict__ b1, const float* __restrict__ w2,
               const float* __restrict__ b2, float* __restrict__ Y)
{
  __shared__ float As[16 * APITCH];
  __shared__ float hid[16 * HPITCH];
  const int tid   = threadIdx.x;
  const int lane  = tid & 31;
  const int wv    = tid >> 5;
  const int lm    = lane & 15;
  const int kh    = lane >> 4;
  const int tileM = blockIdx.x * 16;

  // stage shared X tile once (all 8 waves consume it)
  for (int c = tid; c < 2048; c += 256) {
    const int f = c * 2;
    const int row = f >> 8;
    const int col = f & 255;
    async_ld_b64(&As[row * APITCH + col], X + (size_t)(tileM + row) * DIM_ + col);
  }
  wait_async();
  __syncthreads();

  // phase 1: hidden[16][wv*64 .. wv*64+64) over K=256 from LDS
  v8f acc[4];
#pragma unroll
  for (int t = 0; t < 4; ++t)
#pragma unroll
    for (int r = 0; r < 8; ++r) acc[t][r] = 0.0f;

  {
    const float* Ap = &As[lm * APITCH + kh * 2];
    const float* Bp = w1 + (size_t)(kh * 2) * HID_ + wv * 64 + lm;
#pragma unroll 2
    for (int k = 0; k < DIM_; k += 4) {
      v2f a; a.x = Ap[0]; a.y = Ap[1];
#pragma unroll
      for (int t = 0; t < 4; ++t) {
        v2f b; b.x = Bp[t * 16]; b.y = Bp[HID_ + t * 16];
        acc[t] = wmma4(a, b, acc[t]);
      }
      Ap += 4;
      Bp += (size_t)4 * HID_;
    }
  }
#pragma unroll
  for (int t = 0; t < 4; ++t) {
    const int col = wv * 64 + t * 16 + lm;
    const float bc = b1[col];
#pragma unroll
    for (int r = 0; r < 8; ++r)
      hid[(r + kh * 8) * HPITCH + col] = gelu_f(acc[t][r] + bc);
  }
  __syncthreads();

  // phase 2: out[16][wv*32 .. wv*32+32) over K=512 from LDS
  v8f acc2[2];
#pragma unroll
  for (int t = 0; t < 2; ++t)
#pragma unroll
    for (int r = 0; r < 8; ++r) acc2[t][r] = 0.0f;

  {
    const float* Ap = &hid[lm * HPITCH + kh * 2];
    const float* Bp = w2 + (size_t)(kh * 2) * DIM_ + wv * 32 + lm;
#pragma unroll 2
    for (int k = 0; k < HID_; k += 4) {
      v2f a; a.x = Ap[0]; a.y = Ap[1];
#pragma unroll
      for (int t = 0; t < 2; ++t) {
        v2f b; b.x = Bp[t * 16]; b.y = Bp[DIM_ + t * 16];
        acc2[t] = wmma4(a, b, acc2[t]);
      }
      Ap += 4;
      Bp += (size_t)4 * DIM_;
    }
  }
#pragma unroll
  for (int t = 0; t < 2; ++t) {
    const int col = wv * 32 + t * 16 + lm;
    const float bc = b2[col];
#pragma unroll
    for (int r = 0; r < 8; ++r) {
      const int rloc = r + kh * 8;
      const int row  = tileM + rloc;
      // residual read comes from the LDS-resident X tile
      Y[(size_t)row * DIM_ + col] = As[rloc * APITCH + col] + acc2[t][r] + bc;
    }
  }
}

// ---------------------------------------------------------------------------
// In-place row LayerNorm over 256 columns; one wave per row, 8 rows per block.
// ---------------------------------------------------------------------------
__global__ __launch_bounds__(256)
void ln_inplace(float* __restrict__ X, const float* __restrict__ g,
                const float* __restrict__ b)
{
  const int lane = threadIdx.x & 31;
  const int row  = blockIdx.x * 8 + (threadIdx.x >> 5);
  float* xr = X + (size_t)row * DIM_;
  float vals[8]; float s = 0.f, sq = 0.f;
#pragma unroll
  for (int j = 0; j < 8; ++j) {
    float v = xr[lane + j * 32]; vals[j] = v; s += v; sq += v * v;
  }
  s = wave_sum(s); sq = wave_sum(sq);
  const float mean = s * (1.0f / DIM_);
  const float var  = sq * (1.0f / DIM_) - mean * mean;
  const float inv  = rsqrtf(var + 1e-5f);
#pragma unroll
  for (int j = 0; j < 8; ++j) {
    const int col = lane + j * 32;
    xr[col] = (vals[j] - mean) * inv * g[col] + b[col];
  }
}

// ---------------------------------------------------------------------------
// Edge attention kernels: one wave per edge, lane = head-dim index.
// ---------------------------------------------------------------------------
__global__ __launch_bounds__(256)
void score_self(const float* __restrict__ qkv, const int* __restrict__ src,
                const int* __restrict__ dst, float* __restrict__ logits,
                unsigned* __restrict__ mseg)
{
  const int e = blockIdx.x * 8 + (threadIdx.x >> 5);
  const int lane = threadIdx.x & 31;
  const int s = src[e], d = dst[e];
  const float* qr = qkv + (size_t)d * 768;         // q at offset 0
  const float* kr = qkv + (size_t)s * 768 + 256;   // k at offset 256
#pragma unroll
  for (int hh = 0; hh < H_; ++hh) {
    float p = qr[hh * 32 + lane] * kr[hh * 32 + lane];
    p = wave_sum(p);
    if (lane == 0) {
      const float l = p * SCALE_;
      logits[(size_t)e * H_ + hh] = l;
      atomicMax(&mseg[d * H_ + hh], f2ord(l));
    }
  }
}

__global__ __launch_bounds__(256)
void score_cross(const float* __restrict__ q, const float* __restrict__ kv,
                 const int* __restrict__ dst, float* __restrict__ logits,
                 unsigned* __restrict__ mseg)
{
  const int e = blockIdx.x * 8 + (threadIdx.x >> 5);
  const int lane = threadIdx.x & 31;
  const int d = dst[e];
  const float* qr = q + (size_t)d * DIM_;
  const float* kr = kv + (size_t)e * 512;          // k half of kv row
#pragma unroll
  for (int hh = 0; hh < H_; ++hh) {
    float p = qr[hh * 32 + lane] * kr[hh * 32 + lane];
    p = wave_sum(p);
    if (lane == 0) {
      const float l = p * SCALE_;
      logits[(size_t)e * H_ + hh] = l;
      atomicMax(&mseg[d * H_ + hh], f2ord(l));
    }
  }
}

__global__ __launch_bounds__(256)
void score_cond(const float* __restrict__ qkv2, const float* __restrict__ cbuf,
                const int* __restrict__ src, const int* __restrict__ dst,
                float* __restrict__ score, float* __restrict__ logits,
                unsigned* __restrict__ mseg, float* __restrict__ smean)
{
  const int e = blockIdx.x * 8 + (threadIdx.x >> 5);
  const int lane = threadIdx.x & 31;
  const int s = src[e], d = dst[e];
  const float* qr = qkv2 + (size_t)d * 768;
  const float* kr = qkv2 + (size_t)s * 768 + 256;
  const float* cr = cbuf + (size_t)e * DIM_;
  float* sr = score + (size_t)e * DIM_;
  float macc = 0.f;
#pragma unroll
  for (int hh = 0; hh < H_; ++hh) {
    const float sc = kr[hh * 32 + lane] * qr[hh * 32 + lane] * cr[hh * 32 + lane];
    sr[hh * 32 + lane] = sc;
    macc += sc;
    float p = wave_sum(sc);
    if (lane == 0) {
      const float l = p * SCALE_;
      logits[(size_t)e * H_ + hh] = l;
      atomicMax(&mseg[d * H_ + hh], f2ord(l));
    }
  }
  smean[(size_t)e * HD_ + lane] = macc * 0.125f;   // mean over 8 heads
}

__global__ __launch_bounds__(256)
void exp_norm(float* __restrict__ logits, const unsigned* __restrict__ mseg,
              float* __restrict__ sseg, const int* __restrict__ dst, int total)
{
  const int i = blockIdx.x * 256 + threadIdx.x;
  if (i >= total) return;
  const int e = i >> 3, hh = i & 7;
  const int d = dst[e];
  const float z = expf(logits[i] - ord2f(mseg[d * H_ + hh]));
  logits[i] = z;
  atomicAdd(&sseg[d * H_ + hh], z);
}

__global__ __launch_bounds__(256)
void agg_self(const float* __restrict__ qkv, const float* __restrict__ logits,
              const float* __restrict__ sseg, const int* __restrict__ src,
              const int* __restrict__ dst, float* __restrict__ hnew)
{
  const int e = blockIdx.x * 8 + (threadIdx.x >> 5);
  const int lane = threadIdx.x & 31;
  const int s = src[e], d = dst[e];
  const float* vr = qkv + (size_t)s * 768 + 512;
#pragma unroll
  for (int hh = 0; hh < H_; ++hh) {
    const float w = logits[(size_t)e * H_ + hh] / sseg[d * H_ + hh];
    // swapaxes(1,2) layout: h_sa[n, dd*H + hh]
    atomicAdd(&hnew[(size_t)d * DIM_ + lane * H_ + hh], w * vr[hh * 32 + lane]);
  }
}

__global__ __launch_bounds__(256)
void agg_cross(const float* __restrict__ kv, const float* __restrict__ logits,
               const float* __restrict__ sseg, const int* __restrict__ dst,
               float* __restrict__ hnew)
{
  const int e = blockIdx.x * 8 + (threadIdx.x >> 5);
  const int lane = threadIdx.x & 31;
  const int d = dst[e];
  const float* vr = kv + (size_t)e * 512 + 256;    // v half of kv row
#pragma unroll
  for (int hh = 0; hh < H_; ++hh) {
    const float w = logits[(size_t)e * H_ + hh] / sseg[d * H_ + hh];
    atomicAdd(&hnew[(size_t)d * DIM_ + hh * 32 + lane], w * vr[hh * 32 + lane]);
  }
}

__global__ __launch_bounds__(256)
void agg_cond(const float* __restrict__ qkv2, const float* __restrict__ logits,
              const float* __restrict__ sseg, const int* __restrict__ src,
              const int* __restrict__ dst, float* __restrict__ hnew)
{
  const int e = blockIdx.x * 8 + (threadIdx.x >> 5);
  const int lane = threadIdx.x & 31;
  const int s = src[e], d = dst[e];
  const float* vr = qkv2 + (size_t)s * 768 + 512;
#pragma unroll
  for (int hh = 0; hh < H_; ++hh) {
    const float w = logits[(size_t)e * H_ + hh] / sseg[d * H_ + hh];
    atomicAdd(&hnew[(size_t)d * DIM_ + hh * 32 + lane], w * vr[hh * 32 + lane]);
  }
}

// ---------------------------------------------------------------------------
extern "C" void kernel_launch(void* const* d_in, const int* in_sizes, int n_in,
                              void* d_out, int out_size, void* d_ws, size_t ws_size,
                              hipStream_t stream)
{
  (void)n_in; (void)out_size; (void)ws_size;
  const float* h0 = (const float*)d_in[0];
  const float* e0 = (const float*)d_in[1];
  const int*  src = (const int*)d_in[2];
  const int*  dst = (const int*)d_in[3];
  const float* nsa_qkv_w = (const float*)d_in[4];
  const float* nsa_out_w = (const float*)d_in[5];
  const float* nsa_out_b = (const float*)d_in[6];
  const float* ca_q_w    = (const float*)d_in[7];
  const float* ca_kv_w   = (const float*)d_in[8];
  const float* ca_hp_w   = (const float*)d_in[9];
  const float* ca_hp_b   = (const float*)d_in[10];
  const float* cond_qkv_w= (const float*)d_in[11];
  const float* cond_c_w  = (const float*)d_in[12];
  const float* cond_hp_w = (const float*)d_in[13];
  const float* cond_hp_b = (const float*)d_in[14];
  const float* cond_ep_w = (const float*)d_in[15];
  const float* cond_ep_b = (const float*)d_in[16];
  const float* mlp1_w1 = (const float*)d_in[17];
  const float* mlp1_b1 = (const float*)d_in[18];
  const float* mlp1_w2 = (const float*)d_in[19];
  const float* mlp1_b2 = (const float*)d_in[20];
  const float* mlp2_w1 = (const float*)d_in[21];
  const float* mlp2_b1 = (const float*)d_in[22];
  const float* mlp2_w2 = (const float*)d_in[23];
  const float* mlp2_b2 = (const float*)d_in[24];
  const float* ln1_g  = (const float*)d_in[25];
  const float* ln1_b  = (const float*)d_in[26];
  const float* ln3_g  = (const float*)d_in[27];
  const float* ln3_b  = (const float*)d_in[28];
  const float* ln7_g  = (const float*)d_in[29];
  const float* ln7_b  = (const float*)d_in[30];
  const float* ln8_g  = (const float*)d_in[31];
  const float* ln8_b  = (const float*)d_in[32];
  const float* ln9_g  = (const float*)d_in[33];
  const float* ln9_b  = (const float*)d_in[34];
  const float* ln10_g = (const float*)d_in[35];
  const float* ln10_b = (const float*)d_in[36];

  const int N = in_sizes[0] / DIM_;   // 10000
  const int E = in_sizes[1] / DIM_;   // 250000

  float* out_h  = (float*)d_out;
  float* out_e  = out_h + (size_t)N * DIM_;
  float* out_sm = out_e + (size_t)E * DIM_;

  // workspace layout (floats); score aliases the dead kv region
  float* ws      = (float*)d_ws;
  float* qkv     = ws;                                   // N*768
  float* logits  = qkv + (size_t)N * 768;                // E*8
  unsigned* mseg = (unsigned*)(logits + (size_t)E * 8);  // N*8
  float* sseg    = (float*)(mseg + (size_t)N * 8);       // N*8
  float* hnew    = sseg + (size_t)N * 8;                 // N*256
  float* kvbuf   = hnew + (size_t)N * 256;               // E*512
  float* cbuf    = kvbuf + (size_t)E * 512;              // E*256
  float* scorebuf = kvbuf;                               // alias (kv dead by then)

  const dim3 blk256(256);
  const dim3 gblk128(128), gblk256(256), gblk384(384);   // = (Nc/64)*32
  const int Mh = N / 16, Me = E / 16;
  const int EW = E / 8;
  const int EH = (E * H_ + 255) / 256;
  const int LNh = N / 8, LNe = E / 8;

  auto zero_seg = [&]() {
    (void)hipMemsetAsync(mseg, 0, (size_t)N * H_ * sizeof(unsigned), stream);
    (void)hipMemsetAsync(sseg, 0, (size_t)N * H_ * sizeof(float), stream);
    (void)hipMemsetAsync(hnew, 0, (size_t)N * DIM_ * sizeof(float), stream);
  };

  // ---- Stage A: node self-attention over edges ----
  gemm_wmma_f32<false><<<Mh, gblk384, 0, stream>>>(h0, nullptr, nsa_qkv_w, nullptr, nullptr, qkv, 768, 0);
  zero_seg();
  score_self<<<EW, blk256, 0, stream>>>(qkv, src, dst, logits, mseg);
  exp_norm<<<EH, blk256, 0, stream>>>(logits, mseg, sseg, dst, E * H_);
  agg_self<<<EW, blk256, 0, stream>>>(qkv, logits, sseg, src, dst, hnew);
  // out_h = h0 + (hnew + h0) @ nsa_out_w + b
  gemm_wmma_f32<true><<<Mh, gblk128, 0, stream>>>(hnew, h0, nsa_out_w, nsa_out_b, h0, out_h, 256, 0);
  ln_inplace<<<LNh, blk256, 0, stream>>>(out_h, ln1_g, ln1_b);

  // ---- Stage B: edge->node cross attention ----
  gemm_wmma_f32<false><<<Mh, gblk128, 0, stream>>>(out_h, nullptr, ca_q_w, nullptr, nullptr, qkv, 256, 0);
  gemm_wmma_f32<false><<<Me, gblk256, 0, stream>>>(e0, nullptr, ca_kv_w, nullptr, nullptr, kvbuf, 512, 0);
  zero_seg();
  score_cross<<<EW, blk256, 0, stream>>>(qkv, kvbuf, dst, logits, mseg);
  exp_norm<<<EH, blk256, 0, stream>>>(logits, mseg, sseg, dst, E * H_);
  agg_cross<<<EW, blk256, 0, stream>>>(kvbuf, logits, sseg, dst, hnew);
  gemm_wmma_f32<false><<<Mh, gblk128, 0, stream>>>(hnew, nullptr, ca_hp_w, ca_hp_b, out_h, out_h, 256, 0);
  ln_inplace<<<LNh, blk256, 0, stream>>>(out_h, ln3_g, ln3_b);

  // ---- Stage C: conditional attention ----
  gemm_wmma_f32<false><<<Mh, gblk384, 0, stream>>>(out_h, nullptr, cond_qkv_w, nullptr, nullptr, qkv, 768, 0);
  gemm_wmma_f32<false><<<Me, gblk128, 0, stream>>>(e0, nullptr, cond_c_w, nullptr, nullptr, cbuf, 256, 2); // tanh
  zero_seg();
  score_cond<<<EW, blk256, 0, stream>>>(qkv, cbuf, src, dst, scorebuf, logits, mseg, out_sm);
  exp_norm<<<EH, blk256, 0, stream>>>(logits, mseg, sseg, dst, E * H_);
  agg_cond<<<EW, blk256, 0, stream>>>(qkv, logits, sseg, src, dst, hnew);
  gemm_wmma_f32<false><<<Mh, gblk128, 0, stream>>>(hnew, nullptr, cond_hp_w, cond_hp_b, out_h, out_h, 256, 0);
  ln_inplace<<<LNh, blk256, 0, stream>>>(out_h, ln7_g, ln7_b);
  gemm_wmma_f32<false><<<Me, gblk128, 0, stream>>>(scorebuf, nullptr, cond_ep_w, cond_ep_b, e0, out_e, 256, 0);
  ln_inplace<<<LNe, blk256, 0, stream>>>(out_e, ln8_g, ln8_b);

  // ---- Stage D: MLPs (fused through LDS; hidden never touches HBM) ----
  mlp_fused<<<Mh, blk256, 0, stream>>>(out_h, mlp1_w1, mlp1_b1, mlp1_w2, mlp1_b2, out_h);
  ln_inplace<<<LNh, blk256, 0, stream>>>(out_h, ln9_g, ln9_b);
  mlp_fused<<<Me, blk256, 0, stream>>>(out_e, mlp2_w1, mlp2_b1, mlp2_w2, mlp2_b2, out_e);
  ln_inplace<<<LNe, blk256, 0, stream>>>(out_e, ln10_g, ln10_b);
}